// AttentionNet_68101001445576
// MI455X (gfx1250) — compile-verified
//
#include <hip/hip_runtime.h>
#include <cstdint>

typedef float v2f __attribute__((ext_vector_type(2)));
typedef float v8f __attribute__((ext_vector_type(8)));

static constexpr float NEG_FILL_F = -103.61633455210506f; // log(1e-45)
static constexpr float EXPL_C     = 10.0f;

__device__ __forceinline__ float fast_tanh(float x) {
#if __has_builtin(__builtin_amdgcn_tanhf)
  return __builtin_amdgcn_tanhf(x);
#else
  float r;
  asm("v_tanh_f32 %0, %1" : "=v"(r) : "v"(x));
  return r;
#endif
}

__device__ __forceinline__ float wave_sum(float v) {
#pragma unroll
  for (int o = 16; o > 0; o >>= 1) v += __shfl_xor(v, o, 32);
  return v;
}
__device__ __forceinline__ float wave_max(float v) {
#pragma unroll
  for (int o = 16; o > 0; o >>= 1) v = fmaxf(v, __shfl_xor(v, o, 32));
  return v;
}

// ---------------------------------------------------------------------------
// Fused encode (K=3 or K=6 projection + bias, into LDS) followed by
// out = enc @ Wmat using V_WMMA_F32_16X16X4_F32.
// Block = 256 threads (8 waves). Block owns a 16-row M tile; wave w owns
// the 16-col N tile n0 = 16*w (N = 128). 32 WMMA k-steps cover K = 128.
// ---------------------------------------------------------------------------
__global__ __launch_bounds__(256) void enc_gemm_kernel(
    const float* __restrict__ X, int xstride, int J,
    const float* __restrict__ Wenc,   // (>=J, 128) row-major
    const float* __restrict__ benc,   // (128)
    const float* __restrict__ Wmat,   // (128, 128) row-major
    float* __restrict__ out)          // (M, 128)
{
  __shared__ float encA[16 * 132];    // padded stride: kills bank conflicts
  const int m0 = blockIdx.x * 16;

  // Phase 1: encode 16 rows (tiny K) into LDS.
  for (int i = threadIdx.x; i < 16 * 128; i += 256) {
    const int row = i >> 7, h = i & 127;
    float acc = benc[h];
    const float* xr = X + (size_t)(m0 + row) * xstride;
    for (int j = 0; j < J; ++j) acc = fmaf(xr[j], Wenc[j * 128 + h], acc);
    encA[row * 132 + h] = acc;
  }
  __syncthreads();

  // Phase 2: WMMA f32 16x16x4 accumulation over K=128.
  const int wave = threadIdx.x >> 5;
  const int lane = threadIdx.x & 31;
  const int half = lane >> 4;            // 0: K pair {0,1}; 1: K pair {2,3}
  const int arow = lane & 15;            // A-matrix M row
  const int n    = wave * 16 + (lane & 15);

  v8f C = {};
#pragma unroll 4
  for (int k0 = 0; k0 < 128; k0 += 4) {
    const int ka = k0 + half * 2;
    v2f A, B;
    A.x = encA[arow * 132 + ka];
    A.y = encA[arow * 132 + ka + 1];
    B.x = Wmat[(size_t)ka * 128 + n];
    B.y = Wmat[(size_t)(ka + 1) * 128 + n];
    C = __builtin_amdgcn_wmma_f32_16x16x4_f32(false, A, false, B,
                                              (short)0, C, false, false);
  }

#pragma unroll
  for (int v = 0; v < 8; ++v) {
    const int row = v + half * 8;        // C/D layout: M = vgpr + 8*(lane>=16)
    out[(size_t)(m0 + row) * 128 + n] = C[v];
  }
}

// ---------------------------------------------------------------------------
// Fused tanh-attention + masked softmax.
// Block = 256 threads (8 waves): stages enc_t[b] (256x128 f32 = 128 KB) into
// LDS once; each wave owns one u-row: 256 scores, each a 128-wide
// tanh(enc+dec).vt reduction (4 h per lane + shfl_xor allreduce), then a
// wave-local softmax with coalesced stores.
// ---------------------------------------------------------------------------
__global__ __launch_bounds__(256) void attn_softmax_kernel(
    const float* __restrict__ enc_t,    // (16, 256, 128)
    const float* __restrict__ dec_t,    // (16, 500, 128)
    const uint8_t* __restrict__ masks,  // (16, 500, 256) bool
    const float* __restrict__ vt,       // (128)
    float* __restrict__ probs)          // (16, 500, 256)
{
  __shared__ float enc_s[256 * 128];    // 128 KB of the 320 KB WGP LDS
  const int b = blockIdx.y;

  {
    const float4* src = reinterpret_cast<const float4*>(enc_t + (size_t)b * 256 * 128);
    float4* dst = reinterpret_cast<float4*>(enc_s);
    for (int i = threadIdx.x; i < (256 * 128) / 4; i += 256) dst[i] = src[i];
  }
  __syncthreads();

  const int wave = threadIdx.x >> 5;
  const int lane = threadIdx.x & 31;
  const int u = blockIdx.x * 8 + wave;
  if (u >= 500) return;                 // no WMMA here; divergence is fine

  const size_t row = (size_t)b * 500 + u;
  const float* dec = dec_t + row * 128;
  const float d0 = dec[lane], d1 = dec[lane + 32], d2 = dec[lane + 64], d3 = dec[lane + 96];
  const float v0 = vt[lane],  v1 = vt[lane + 32],  v2 = vt[lane + 64],  v3 = vt[lane + 96];
  const uint8_t* mrow = masks + row * 256;

  bool mk[8];
#pragma unroll
  for (int j = 0; j < 8; ++j) mk[j] = mrow[j * 32 + lane] != 0;

  float sc[8];
#pragma unroll
  for (int j = 0; j < 8; ++j) {         // unrolled: sc[] stays in registers
    float keep = 0.0f;
    for (int s32 = 0; s32 < 32; ++s32) {
      const float* er = enc_s + (j * 32 + s32) * 128;
      float p = fast_tanh(er[lane]      + d0) * v0;
      p      += fast_tanh(er[lane + 32] + d1) * v1;
      p      += fast_tanh(er[lane + 64] + d2) * v2;
      p      += fast_tanh(er[lane + 96] + d3) * v3;
      p = wave_sum(p);                  // uniform across the wave
      if (s32 == lane) keep = p;        // lane keeps s = j*32 + lane
    }
    sc[j] = (mk[j] ? keep : NEG_FILL_F) * EXPL_C;
  }

  float m = sc[0];
#pragma unroll
  for (int j = 1; j < 8; ++j) m = fmaxf(m, sc[j]);
  m = wave_max(m);

  float sum = 0.0f;
#pragma unroll
  for (int j = 0; j < 8; ++j) { sc[j] = __expf(sc[j] - m); sum += sc[j]; }
  sum = wave_sum(sum);
  const float inv = 1.0f / sum;

  float* orow = probs + row * 256;
#pragma unroll
  for (int j = 0; j < 8; ++j) orow[j * 32 + lane] = sc[j] * inv;  // coalesced
}

// ---------------------------------------------------------------------------
extern "C" void kernel_launch(void* const* d_in, const int* in_sizes, int n_in,
                              void* d_out, int out_size, void* d_ws, size_t ws_size,
                              hipStream_t stream) {
  (void)in_sizes; (void)n_in; (void)out_size; (void)ws_size;
  const float*   user_in   = (const float*)d_in[0];   // (16,500,4)
  const float*   server_in = (const float*)d_in[1];   // (16,256,6)
  const uint8_t* masks     = (const uint8_t*)d_in[2]; // (16,500,256) bool
  const float*   Wu        = (const float*)d_in[3];   // (3,128)
  const float*   bu        = (const float*)d_in[4];   // (128)
  const float*   Ws        = (const float*)d_in[5];   // (7,128) row 6 unused (active==0)
  const float*   bs        = (const float*)d_in[6];   // (128)
  const float*   W1        = (const float*)d_in[7];   // (128,128)
  const float*   W2        = (const float*)d_in[8];   // (128,128)
  const float*   vt        = (const float*)d_in[9];   // (128)
  float*         probs     = (float*)d_out;           // (16,500,256)

  float* dec_t = (float*)d_ws;                        // 8000*128 f32 = 4 MB
  float* enc_t = dec_t + (size_t)8000 * 128;          // 4096*128 f32 = 2 MB

  // dec_t = (user[...,:3] @ Wu + bu) @ W2   : M = 8000 -> 500 tiles
  enc_gemm_kernel<<<dim3(500), 256, 0, stream>>>(user_in, 4, 3, Wu, bu, W2, dec_t);
  // enc_t = (server ++ 0 @ Ws + bs) @ W1    : M = 4096 -> 256 tiles
  enc_gemm_kernel<<<dim3(256), 256, 0, stream>>>(server_in, 6, 6, Ws, bs, W1, enc_t);
  // fused tanh-attention + masked softmax
  attn_softmax_kernel<<<dim3(63, 16), 256, 0, stream>>>(enc_t, dec_t, masks, vt, probs);
}